// AttentionLayerSparse_20177756356658
// MI455X (gfx1250) — compile-verified
//
#include <hip/hip_runtime.h>
#include <hip/hip_bf16.h>
#include <math.h>

typedef float v2f __attribute__((ext_vector_type(2)));
typedef float v8f __attribute__((ext_vector_type(8)));

#define N_NODES 20000
#define N_EDGES 640000
#define FIN_DIM 256
#define FQK_DIM 256

// ---------------------------------------------------------------------------
// Zero a float buffer (expsum accumulator in workspace).
// ---------------------------------------------------------------------------
__global__ __launch_bounds__(256) void zero_f32(float* __restrict__ p, int n) {
  int i = blockIdx.x * 256 + threadIdx.x;
  if (i < n) p[i] = 0.0f;
}

// ---------------------------------------------------------------------------
// qk = x @ W^T  (M=20000, N=512, K=256, fp32) via V_WMMA_F32_16X16X4_F32.
// Each wave computes a 16x64 strip: 4 accumulators, A-fragment reused 4x.
// Columns [0,256) -> q (scaled by FQK^-0.5 = 1/16), [256,512) -> k.
//
// Fragment layouts per CDNA5 ISA 7.12.2 (wave32):
//   A 16x4 f32 : lane&15 = M row; lane>>4 selects K pair {0,1} or {2,3}.
//   B 4x16 f32 : lane&15 = N col; lane>>4 selects K pair (symmetric).
//   C/D 16x16  : VGPR i holds row i (lanes 0-15) / row 8+i (lanes 16-31).
// ---------------------------------------------------------------------------
__global__ __launch_bounds__(256) void gemm_qk(const float* __restrict__ x,
                                               const float* __restrict__ W,
                                               float* __restrict__ q,
                                               float* __restrict__ kmat) {
  const int lane  = threadIdx.x & 31;
  const int strip = blockIdx.x * 8 + (threadIdx.x >> 5); // 1250*8 = 10000 strips
  const int mBase = (strip >> 3) << 4;                   // 16-row tile
  const int nBase = (strip & 7) << 6;                    // 64-col strip
  const int mRow  = lane & 15;
  const int nCol  = lane & 15;
  const int kHalf = (lane >> 4) << 1;                    // 0 or 2

  const float* __restrict__ xp = x + (size_t)(mBase + mRow) * FIN_DIM + kHalf;
  const float* __restrict__ w0 = W + (size_t)(nBase + nCol) * FIN_DIM + kHalf;
  const float* __restrict__ w1 = w0 + 16 * FIN_DIM;
  const float* __restrict__ w2 = w0 + 32 * FIN_DIM;
  const float* __restrict__ w3 = w0 + 48 * FIN_DIM;

  v8f c0 = {}, c1 = {}, c2 = {}, c3 = {};

#pragma unroll 4
  for (int k0 = 0; k0 < FIN_DIM; k0 += 4) {
    v2f a  = *(const v2f*)(xp + k0);
    v2f b0 = *(const v2f*)(w0 + k0);
    v2f b1 = *(const v2f*)(w1 + k0);
    v2f b2 = *(const v2f*)(w2 + k0);
    v2f b3 = *(const v2f*)(w3 + k0);
    c0 = __builtin_amdgcn_wmma_f32_16x16x4_f32(false, a, false, b0, (short)0, c0, false, false);
    c1 = __builtin_amdgcn_wmma_f32_16x16x4_f32(false, a, false, b1, (short)0, c1, false, false);
    c2 = __builtin_amdgcn_wmma_f32_16x16x4_f32(false, a, false, b2, (short)0, c2, false, false);
    c3 = __builtin_amdgcn_wmma_f32_16x16x4_f32(false, a, false, b3, (short)0, c3, false, false);
  }

  const int mOff = (lane >> 4) ? 8 : 0;
  v8f cc[4] = {c0, c1, c2, c3};

#pragma unroll
  for (int j = 0; j < 4; ++j) {
    const int col = nBase + j * 16 + nCol;
    float* __restrict__ dst;
    float scale;
    if (col < FQK_DIM) { dst = q + col;                 scale = 0.0625f; } // FQK^-0.5
    else               { dst = kmat + (col - FQK_DIM);  scale = 1.0f;    }
#pragma unroll
    for (int i = 0; i < 8; ++i) {
      const int m = mBase + mOff + i;
      dst[(size_t)m * FQK_DIM] = cc[j][i] * scale;
    }
  }
}

// ---------------------------------------------------------------------------
// Per-edge: aw = q[src] . k[dest] (256-dim), ex = exp(aw) -> d_out,
// atomic-accumulate ex into expsum[src]. One wave32 per edge; each lane
// handles 8 features via two float4 (b128) loads per matrix. q/k (41 MB)
// live in L2, so gathers stream from L2 at cache bandwidth.
// ---------------------------------------------------------------------------
__global__ __launch_bounds__(256) void edge_exp(const float* __restrict__ q,
                                                const float* __restrict__ kmat,
                                                const int* __restrict__ src,
                                                const int* __restrict__ dst,
                                                float* __restrict__ ex,
                                                float* __restrict__ expsum,
                                                int nE) {
  const int gid  = blockIdx.x * 256 + threadIdx.x;
  const int e    = gid >> 5;
  const int lane = threadIdx.x & 31;
  if (e >= nE) return;

  const int s = src[e];
  const int d = dst[e];
  const float4* __restrict__ qp = (const float4*)(q    + (size_t)s * FQK_DIM);
  const float4* __restrict__ kp = (const float4*)(kmat + (size_t)d * FQK_DIM);

  const float4 qa = qp[lane];      const float4 ka = kp[lane];
  const float4 qb = qp[lane + 32]; const float4 kb = kp[lane + 32];

  float sum = qa.x * ka.x + qa.y * ka.y + qa.z * ka.z + qa.w * ka.w
            + qb.x * kb.x + qb.y * kb.y + qb.z * kb.z + qb.w * kb.w;

  sum += __shfl_xor(sum, 16, 32);
  sum += __shfl_xor(sum, 8, 32);
  sum += __shfl_xor(sum, 4, 32);
  sum += __shfl_xor(sum, 2, 32);
  sum += __shfl_xor(sum, 1, 32);

  if (lane == 0) {
    const float v = __expf(sum);
    ex[e] = v;
    unsafeAtomicAdd(expsum + s, v);   // global_atomic_add_f32
  }
}

// ---------------------------------------------------------------------------
// out[e] = ex[e] / expsum[src[e]]
// ---------------------------------------------------------------------------
__global__ __launch_bounds__(256) void edge_norm(float* __restrict__ ex,
                                                 const int* __restrict__ src,
                                                 const float* __restrict__ expsum,
                                                 int nE) {
  const int e = blockIdx.x * 256 + threadIdx.x;
  if (e < nE) ex[e] = ex[e] / expsum[src[e]];
}

// ---------------------------------------------------------------------------
extern "C" void kernel_launch(void* const* d_in, const int* in_sizes, int n_in,
                              void* d_out, int out_size, void* d_ws, size_t ws_size,
                              hipStream_t stream) {
  const float* x  = (const float*)d_in[0];
  const int*   ei = (const int*)d_in[1];   // [2, E] int32
  const float* W  = (const float*)d_in[2]; // [512, 256]
  float* out = (float*)d_out;              // [E]

  // Workspace layout: q [N,256] | k [N,256] | expsum [N]
  float* q      = (float*)d_ws;
  float* kmat   = q + (size_t)N_NODES * FQK_DIM;
  float* expsum = kmat + (size_t)N_NODES * FQK_DIM;

  const int* src = ei;
  const int* dst = ei + N_EDGES;

  zero_f32<<<(N_NODES + 255) / 256, 256, 0, stream>>>(expsum, N_NODES);
  gemm_qk<<<1250, 256, 0, stream>>>(x, W, q, kmat);                 // 10000 strips / 8 waves per block
  edge_exp<<<N_EDGES / 8, 256, 0, stream>>>(q, kmat, src, dst, out, expsum, N_EDGES);
  edge_norm<<<(N_EDGES + 255) / 256, 256, 0, stream>>>(out, src, expsum, N_EDGES);
}